// SequenceAtt_17944373363196
// MI455X (gfx1250) — compile-verified
//
#include <hip/hip_runtime.h>
#include <hip/hip_bf16.h>

typedef float v2f __attribute__((ext_vector_type(2)));
typedef float v8f __attribute__((ext_vector_type(8)));
typedef int   v4i __attribute__((ext_vector_type(4)));

#define T_LEN 2048
#define HID   512
#define G4H   2048          // 4*HID
#define WSTR  516           // padded LDS row stride (floats): bank-conflict fix

// Async global->LDS path (CDNA5 ASYNCcnt), guarded so a missing builtin
// degrades to plain loads instead of breaking the build.
#if defined(__has_builtin)
#if __has_builtin(__builtin_amdgcn_global_load_async_to_lds_b128) && \
    __has_builtin(__builtin_amdgcn_s_wait_asynccnt)
#define HAVE_ASYNC_LDS 1
#endif
#endif
#ifndef HAVE_ASYNC_LDS
#define HAVE_ASYNC_LDS 0
#endif

#if HAVE_ASYNC_LDS
typedef __attribute__((address_space(1))) v4i gv4i_t;   // global int4*
typedef __attribute__((address_space(3))) v4i lv4i_t;   // LDS int4*
#endif

// ---------------------------------------------------------------------------
// GEMM: out[t, r] = sum_k A[t,k] * W[r,k] + (bih[r] + bhh[r])
// A: [2048 x K], W: [2048 x K] row-major, out: [2048 x 2048].
// One wave computes TWO 16x16 tiles (shared A fragment, independent WMMA
// accumulator chains) via V_WMMA_F32_16X16X4_F32 (fp32 exact).
// ---------------------------------------------------------------------------
__global__ void __launch_bounds__(128)
gemm_inproj_wmma(const float* __restrict__ A,
                 const float* __restrict__ W,
                 const float* __restrict__ bih,
                 const float* __restrict__ bhh,
                 float* __restrict__ out, int K) {
    const int wave  = threadIdx.x >> 5;
    const int lane  = threadIdx.x & 31;
    const int pair  = (blockIdx.x & 15) * 4 + wave;   // 0..63 (N tile-pairs)
    const int tileM = blockIdx.x >> 4;                // 0..127
    const int m0 = tileM * 16;
    const int n0 = pair * 32;
    const int lrow  = lane & 15;
    const int khalf = lane >> 4;                      // 0: k={0,1}  1: k={2,3}

    const float* arow  = A + (size_t)(m0 + lrow) * K + 2 * khalf;
    const float* brow0 = W + (size_t)(n0 + lrow) * K + 2 * khalf;
    const float* brow1 = W + (size_t)(n0 + 16 + lrow) * K + 2 * khalf;

    v8f acc0 = {}, acc1 = {};
    for (int k = 0; k < K; k += 4) {
        v2f a  = *(const v2f*)(arow + k);
        v2f b0 = *(const v2f*)(brow0 + k);
        v2f b1 = *(const v2f*)(brow1 + k);
        acc0 = __builtin_amdgcn_wmma_f32_16x16x4_f32(
            false, a, false, b0, (short)0, acc0, false, false);
        acc1 = __builtin_amdgcn_wmma_f32_16x16x4_f32(
            false, a, false, b1, (short)0, acc1, false, false);
    }

    const int col0 = n0 + lrow;
    const int col1 = n0 + 16 + lrow;
    const float bs0 = bih[col0] + bhh[col0];
    const float bs1 = bih[col1] + bhh[col1];
    #pragma unroll
    for (int v = 0; v < 8; ++v) {
        const int rowt = m0 + v + 8 * khalf;          // D layout: lanes16-31 -> M=8+v
        out[(size_t)rowt * G4H + col0] = acc0[v] + bs0;
        out[(size_t)rowt * G4H + col1] = acc1[v] + bs1;
    }
}

// ---------------------------------------------------------------------------
// Reset cross-WG h exchange buffers + sync counter (before each layer).
// ---------------------------------------------------------------------------
__global__ void rec_init(float* hglob, unsigned* cnt) {
    for (int i = threadIdx.x; i < 2 * HID; i += blockDim.x) hglob[i] = 0.0f;
    if (threadIdx.x == 0) *cnt = 0u;
}

// ---------------------------------------------------------------------------
// Persistent recurrence: 16 workgroups x 256 threads. WG b owns hidden units
// j in [b*32, b*32+32) -> the 128 rows {g*512 + b*32 + jl} of W_hh, staged in
// ~258KB of LDS with a padded row stride (WSTR=516 floats) so the split-k
// ds_load_b128 stream is ~conflict-free. Per step: LDS matvec, gate math on
// the 32 owner lanes, h broadcast via double-buffered global vector with a
// release-add / acquire-spin counter.
// ---------------------------------------------------------------------------
__global__ void __launch_bounds__(256)
lstm_recurrent(const float* __restrict__ w_hh,    // [2048 x 512]
               const float* __restrict__ xproj,   // [2048 x 2048], bias fused
               float* __restrict__ hout,          // [2048 x 512]
               float* __restrict__ hglob,         // [2 x 512] double buffer
               unsigned* __restrict__ cnt) {
    extern __shared__ float sh[];
    float* Wsh     = sh;                    // 128 * WSTR
    float* h_sh    = Wsh + 128 * WSTR;      // 512
    float* part    = h_sh + HID;            // 256
    float* gate_sh = part + 256;            // 128
    float* c_sh    = gate_sh + 128;         // 32

    const int tid = threadIdx.x;
    const int b   = blockIdx.x;             // 0..15

    // Stage this WG's 128 rows of W_hh into LDS in float4 granules.
    // (async path when available: GLOBAL_LOAD_ASYNC_TO_LDS_B128 + ASYNCcnt)
    for (int i = tid; i < 128 * (HID / 4); i += 256) {
        const int rl = i >> 7;              // local row 0..127 (128 f4/row)
        const int kq = i & 127;
        const int gate = rl >> 5, jl = rl & 31;
        const int rg = gate * HID + b * 32 + jl;
        const float* src = w_hh + (size_t)rg * HID + kq * 4;
        float* dst = Wsh + rl * WSTR + kq * 4;
#if HAVE_ASYNC_LDS
        __builtin_amdgcn_global_load_async_to_lds_b128(
            (gv4i_t*)src, (lv4i_t*)dst, 0, 0);
#else
        *(float4*)dst = *(const float4*)src;
#endif
    }
#if HAVE_ASYNC_LDS
    __builtin_amdgcn_s_wait_asynccnt(0);
#endif
    if (tid < 32) c_sh[tid] = 0.0f;
    __syncthreads();

    const int row   = tid & 127;            // local gate-row
    const int khalf = tid >> 7;             // split-k half

    for (int t = 0; t < T_LEN; ++t) {
        // Wait until every WG published h_{t-1}.
        if (tid == 0) {
            while (__hip_atomic_load(cnt, __ATOMIC_ACQUIRE,
                                     __HIP_MEMORY_SCOPE_AGENT) < (unsigned)(16 * t)) {
                __builtin_amdgcn_s_sleep(1);
            }
        }
        __syncthreads();

        // h_{t-1} lives in buffer (t-1)&1 == (t&1)^1 (zeroed for t=0).
        const float* hprev = hglob + ((t & 1) ^ 1) * HID;
        for (int i = tid; i < HID; i += 256) h_sh[i] = hprev[i];
        __syncthreads();

        // Partial dot: 2 lanes per row, 256 k each, b128 LDS reads.
        const float4* wr = (const float4*)(Wsh + row * WSTR + khalf * 256);
        const float4* hv = (const float4*)(h_sh + khalf * 256);
        float acc = 0.0f;
        #pragma unroll 8
        for (int k = 0; k < 64; ++k) {
            const float4 w4 = wr[k];
            const float4 h4 = hv[k];      // same addr across lanes: broadcast
            acc += w4.x * h4.x + w4.y * h4.y + w4.z * h4.z + w4.w * h4.w;
        }
        part[tid] = acc;
        __syncthreads();

        if (tid < 128) {
            const int gate = row >> 5, jl = row & 31;
            const int rg = gate * HID + b * 32 + jl;
            gate_sh[row] = part[row] + part[row + 128] + xproj[(size_t)t * G4H + rg];
            if (t + 1 < T_LEN)                        // global_prefetch_b8
                __builtin_prefetch(xproj + (size_t)(t + 1) * G4H + rg, 0, 1);
        }
        __syncthreads();

        if (tid < 32) {
            const float gi = gate_sh[tid];
            const float gf = gate_sh[32 + tid];
            const float gg = gate_sh[64 + tid];
            const float go = gate_sh[96 + tid];
            const float i_ = 1.0f / (1.0f + __expf(-gi));
            const float f_ = 1.0f / (1.0f + __expf(-gf));
            const float g_ = tanhf(gg);
            const float o_ = 1.0f / (1.0f + __expf(-go));
            const float c  = f_ * c_sh[tid] + i_ * g_;
            c_sh[tid] = c;
            const float h = o_ * tanhf(c);
            const int j = b * 32 + tid;
            hglob[(t & 1) * HID + j] = h;            // publish for all WGs
            hout[(size_t)t * HID + j] = h;           // layer output
        }
        __syncthreads();

        if (tid == 0) {
            __threadfence();                         // device-scope visibility
            __hip_atomic_fetch_add(cnt, 1u, __ATOMIC_RELEASE,
                                   __HIP_MEMORY_SCOPE_AGENT);
        }
    }
}

// ---------------------------------------------------------------------------
// out[j] = (1/T) * sum_t softmax_t(tanh(h_t . w_lin + b_lin)) * h[t][j]
// ---------------------------------------------------------------------------
__global__ void __launch_bounds__(1024)
attn_pool(const float* __restrict__ h,        // [2048 x 512]
          const float* __restrict__ w_lin,    // [512]
          const float* __restrict__ b_lin,    // [1]
          float* __restrict__ out) {          // [512]
    __shared__ float wsh[HID];
    __shared__ float e[T_LEN];
    __shared__ float rbuf[1024];
    const int tid = threadIdx.x;

    for (int i = tid; i < HID; i += 1024) wsh[i] = w_lin[i];
    __syncthreads();

    const float bl = b_lin[0];
    for (int t = tid; t < T_LEN; t += 1024) {
        const float4* hr = (const float4*)(h + (size_t)t * HID);
        const float4* wv = (const float4*)wsh;
        float d = 0.0f;
        #pragma unroll 8
        for (int k = 0; k < HID / 4; ++k) {
            const float4 a = hr[k], w4 = wv[k];
            d += a.x * w4.x + a.y * w4.y + a.z * w4.z + a.w * w4.w;
        }
        e[t] = tanhf(d + bl);
    }
    __syncthreads();

    float m = -1e30f;
    for (int t = tid; t < T_LEN; t += 1024) m = fmaxf(m, e[t]);
    rbuf[tid] = m;
    __syncthreads();
    for (int s = 512; s > 0; s >>= 1) {
        if (tid < s) rbuf[tid] = fmaxf(rbuf[tid], rbuf[tid + s]);
        __syncthreads();
    }
    const float mx = rbuf[0];
    __syncthreads();

    float lsum = 0.0f;
    for (int t = tid; t < T_LEN; t += 1024) {
        const float p = __expf(e[t] - mx);
        e[t] = p;
        lsum += p;
    }
    rbuf[tid] = lsum;
    __syncthreads();
    for (int s = 512; s > 0; s >>= 1) {
        if (tid < s) rbuf[tid] += rbuf[tid + s];
        __syncthreads();
    }
    const float inv = 1.0f / (rbuf[0] * (float)T_LEN);   // softmax norm * mean
    __syncthreads();

    for (int j = tid; j < HID; j += 1024) {
        float a = 0.0f;
        for (int t = 0; t < T_LEN; ++t) a += e[t] * h[(size_t)t * HID + j];
        out[j] = a * inv;
    }
}

// ---------------------------------------------------------------------------
extern "C" void kernel_launch(void* const* d_in, const int* in_sizes, int n_in,
                              void* d_out, int out_size, void* d_ws, size_t ws_size,
                              hipStream_t stream) {
    (void)in_sizes; (void)n_in; (void)out_size; (void)ws_size;
    const float* x     = (const float*)d_in[0];
    const float* w_ih0 = (const float*)d_in[1];
    const float* w_hh0 = (const float*)d_in[2];
    const float* b_ih0 = (const float*)d_in[3];
    const float* b_hh0 = (const float*)d_in[4];
    const float* w_ih  = (const float*)d_in[5];   // [4, 2048, 512]
    const float* w_hh  = (const float*)d_in[6];   // [4, 2048, 512]
    const float* b_ih  = (const float*)d_in[7];   // [4, 2048]
    const float* b_hh  = (const float*)d_in[8];   // [4, 2048]
    const float* w_lin = (const float*)d_in[9];
    const float* b_lin = (const float*)d_in[10];

    float* ws      = (float*)d_ws;
    float* xproj   = ws;                                  // 2048*2048
    float* hA      = xproj + (size_t)T_LEN * G4H;         // 2048*512
    float* hB      = hA    + (size_t)T_LEN * HID;         // 2048*512
    float* hglob   = hB    + (size_t)T_LEN * HID;         // 2*512
    unsigned* cnt  = (unsigned*)(hglob + 2 * HID);

    const size_t shbytes =
        (size_t)(128 * WSTR + HID + 256 + 128 + 32) * sizeof(float);  // ~262KB
    const dim3 gemm_grid(2048), gemm_blk(128);

    // Layer 0 (K = 64)
    rec_init<<<1, 256, 0, stream>>>(hglob, cnt);
    gemm_inproj_wmma<<<gemm_grid, gemm_blk, 0, stream>>>(x, w_ih0, b_ih0, b_hh0, xproj, 64);
    lstm_recurrent<<<16, 256, shbytes, stream>>>(w_hh0, xproj, hA, hglob, cnt);

    const float* hin = hA;
    float* hout = hB;
    for (int l = 0; l < 4; ++l) {
        rec_init<<<1, 256, 0, stream>>>(hglob, cnt);
        gemm_inproj_wmma<<<gemm_grid, gemm_blk, 0, stream>>>(
            hin, w_ih + (size_t)l * G4H * HID,
            b_ih + (size_t)l * G4H, b_hh + (size_t)l * G4H, xproj, HID);
        lstm_recurrent<<<16, 256, shbytes, stream>>>(
            w_hh + (size_t)l * G4H * HID, xproj, hout, hglob, cnt);
        float* tmp = (float*)hin; hin = hout; hout = tmp;
    }

    attn_pool<<<1, 1024, 0, stream>>>(hin, w_lin, b_lin, (float*)d_out);
}